// NEC_38259568673027
// MI455X (gfx1250) — compile-verified
//
#include <hip/hip_runtime.h>
#include <hip/hip_bf16.h>

// Problem sizes (fixed by the reference setup)
#define B_SZ   256
#define OBS_SZ 4096
#define D_SZ   128
#define A_SZ   16
#define C_SZ   65536
#define P_SZ   50
#define DELTA_K 1e-3f

#define BT 16            // batch rows per workgroup
#define NC 64            // key columns (C dim) per chunk
#define KT_PAD 132       // LDS row pitch (floats); TDM pad reproduces this
#define D2_PAD 68

typedef float v2f __attribute__((ext_vector_type(2)));
typedef float v4f __attribute__((ext_vector_type(4)));
typedef float v8f __attribute__((ext_vector_type(8)));
typedef unsigned int u32x4 __attribute__((ext_vector_type(4)));
typedef int i32x4 __attribute__((ext_vector_type(4)));
typedef int i32x8 __attribute__((ext_vector_type(8)));

static __device__ __forceinline__ int imin(int a, int b) { return a < b ? a : b; }

// ---------------------------------------------------------------------------
// Kernel 1: h = obs @ W + b  and  hn[b] = sum_d h[b,d]^2
// ---------------------------------------------------------------------------
__global__ __launch_bounds__(128) void nec_embed(const float* __restrict__ obs,
                                                 const float* __restrict__ W,
                                                 const float* __restrict__ bias,
                                                 float* __restrict__ h,
                                                 float* __restrict__ hn) {
  const int b = blockIdx.x;
  const int d = threadIdx.x;  // 0..127
  const float* orow = obs + (size_t)b * OBS_SZ;
  float acc = 0.f;
#pragma unroll 8
  for (int k = 0; k < OBS_SZ; ++k) {
    acc = __builtin_fmaf(orow[k], W[(size_t)k * D_SZ + d], acc);
  }
  const float hv = acc + bias[d];
  h[(size_t)b * D_SZ + d] = hv;

  __shared__ float red[128];
  red[d] = hv * hv;
  __syncthreads();
#pragma unroll
  for (int s = 64; s > 0; s >>= 1) {
    if (d < s) red[d] += red[d + s];
    __syncthreads();
  }
  if (d == 0) hn[b] = red[0];
}

// ---------------------------------------------------------------------------
// TDM: one tensor_load_to_lds moves a 64x128-f32 key tile (rows of 512B) into
// LDS with hardware padding of 4 DWORDs after every 128 DWORDs, reproducing
// the 132-float (528B) bank-conflict-free pitch. Wave-uniform; TENSORcnt.
//   group0: count=1 | lds_addr | global_addr[56:0] | type=2
//   group1: data_size=4B, pad_enable, pad_interval=6 (128 DW), pad_amount=3
//           (4 DW), tensor_dim0=128, tensor_dim1=65536, tile_dim0=128,
//           tile_dim1=64, tensor_dim0_stride=128
// This toolchain exposes the 6-argument builtin form (extra i32x8 group
// before cpol) -- pass zero-filled groups for the unused 2D case.
// ---------------------------------------------------------------------------
static __device__ __forceinline__ void tdm_load_chunk(unsigned long long gaddr,
                                                      unsigned lds_addr) {
  u32x4 g0;
  g0.x = 1u;                                    // count=1 (valid descriptor)
  g0.y = lds_addr;                              // LDS byte address
  g0.z = (unsigned)gaddr;                       // global_addr[31:0]
  g0.w = (unsigned)((gaddr >> 32) & 0x01FFFFFFull) | (2u << 30);  // [56:32]|type=2

  i32x8 g1;
  g1[0] = (int)((2u << 16) | (1u << 20) | (6u << 22) | (3u << 25));
  g1[1] = (int)(128u << 16);        // tensor_dim0[15:0]=128 at bits 63:48
  g1[2] = 0;                        // tensor_dim0 hi, tensor_dim1[15:0]=0
  g1[3] = (int)(1u | (128u << 16)); // tensor_dim1[31:16]=1 (65536), tile_dim0=128
  g1[4] = 64;                       // tile_dim1=64, tile_dim2=0
  g1[5] = 128;                      // tensor_dim0_stride[31:0]=128 elements
  g1[6] = 0;
  g1[7] = 0;

  i32x4 gz4 = {0, 0, 0, 0};             // groups 2/3: unused dims (2D tile)
  i32x8 gz8 = {0, 0, 0, 0, 0, 0, 0, 0}; // extra group (6-arg builtin form)
  __builtin_amdgcn_tensor_load_to_lds(g0, g1, gz4, gz4, gz8, 0);
}

// ---------------------------------------------------------------------------
// Kernel 2: fused distances (WMMA f32) + streaming top-50 + kernel-weighted Q
// grid = (A_SZ, B_SZ/BT), block = 128 threads = 4 waves (wave32)
//   wave w computes the 16x16 dots tile for key columns [16w, 16w+16)
//   quarter-wave (8 lanes) owns one batch row for top-k maintenance/merge
// Key chunks are double-buffered in LDS, DMA'd by the Tensor Data Mover
// (issued by wave 0), overlapping the WMMA compute of the resident chunk.
// The WMMA A-operand (h tile) is hoisted into registers once.
// ---------------------------------------------------------------------------
__global__ __launch_bounds__(128) void nec_topk(const float* __restrict__ mem_keys,
                                                const float* __restrict__ mem_values,
                                                const float* __restrict__ h,
                                                const float* __restrict__ hn,
                                                float* __restrict__ out) {
  const int a     = blockIdx.x;
  const int brow0 = blockIdx.y * BT;
  const int t     = threadIdx.x;   // 0..127
  const int lane  = t & 31;
  const int wave  = t >> 5;

  __shared__ float kt[2][NC][KT_PAD];  // double-buffered key chunk, [c][d]
  __shared__ float lh[BT][KT_PAD];     // h tile, [m][d], padded pitch
  __shared__ float d2buf[BT][D2_PAD];  // d2 tile for current chunk
  __shared__ float mn_s[NC];           // ||k||^2 for chunk columns
  __shared__ float hn_s[BT];           // ||h||^2 for tile rows

  // LDS byte offsets of the two key buffers (flat shared addr truncates to LDS)
  const unsigned kt_lds0 = (unsigned)(size_t)(&kt[0][0][0]);
  const unsigned kt_lds1 = (unsigned)(size_t)(&kt[1][0][0]);
  const unsigned long long kgbase =
      (unsigned long long)(size_t)(mem_keys + ((size_t)a * C_SZ) * D_SZ);

  // --- load h tile (16 x 128) and hn once ---
  {
    const int r  = t >> 3;          // 0..15
    const int cc = (t & 7) * 16;    // 16 floats per thread
    const float* src = h + (size_t)(brow0 + r) * D_SZ + cc;
#pragma unroll
    for (int i = 0; i < 4; ++i)
      ((v4f*)&lh[r][cc])[i] = ((const v4f*)src)[i];
  }
  if (t < BT) hn_s[t] = hn[brow0 + t];

  // --- private streaming top-50 state (scratch) ---
  float nd[P_SZ];
  int   ni[P_SZ];
#pragma unroll
  for (int i = 0; i < P_SZ; ++i) { nd[i] = __builtin_inff(); ni[i] = 0; }
  float worst    = __builtin_inff();
  int   worstpos = 0;

  const int srow  = wave * 4 + (lane >> 3);  // 0..15: batch row owned for top-k
  const int sslot = lane & 7;                // candidate column class (mod 8)

  // prologue: TDM fill of chunk 0 into buffer 0 (wave 0 drives the DMA)
  if (wave == 0) tdm_load_chunk(kgbase, kt_lds0);
  int buf = 0;

  __syncthreads();

  // --- hoist WMMA A-operand (invariant across all chunks) into registers ---
  const int ml = lane & 15;         // A: row M / B: col N
  const int kg = (lane >> 4) * 2;   // K sub-offset per lane half (ISA layout)
  v2f areg[D_SZ / 4];
#pragma unroll
  for (int s = 0; s < D_SZ / 4; ++s)
    areg[s] = *(const v2f*)&lh[ml][4 * s + kg];

  // =============================== C stream ===============================
  for (int c0 = 0; c0 < C_SZ; c0 += NC) {
    // overlap: wave 0 kicks the next chunk's TDM into the other buffer, then
    // waits for the current chunk's DMA; everyone else just hits the barrier.
    if (wave == 0) {
      if (c0 + NC < C_SZ) {
        tdm_load_chunk(kgbase + (unsigned long long)(c0 + NC) * (D_SZ * 4),
                       buf ? kt_lds0 : kt_lds1);
        __builtin_amdgcn_s_wait_tensorcnt(1);
      } else {
        __builtin_amdgcn_s_wait_tensorcnt(0);
      }
    }
    __syncthreads();  // current buffer resident for all waves

    const float (*ktc)[KT_PAD] = kt[buf];

    // per-column squared norms for this chunk
    if (t < NC) {
      float s = 0.f;
#pragma unroll
      for (int d = 0; d < D_SZ; d += 4) {
        v4f kv = *(const v4f*)&ktc[t][d];
        s = __builtin_fmaf(kv.x, kv.x, s);
        s = __builtin_fmaf(kv.y, kv.y, s);
        s = __builtin_fmaf(kv.z, kv.z, s);
        s = __builtin_fmaf(kv.w, kv.w, s);
      }
      mn_s[t] = s;
    }
    __syncthreads();

    // WMMA 16x16 dots tile per wave (A from registers, B from LDS), then d2
    {
      const int nt = wave;              // n-tile within chunk
      v8f acc = {0.f, 0.f, 0.f, 0.f, 0.f, 0.f, 0.f, 0.f};
#pragma unroll
      for (int s = 0; s < D_SZ / 4; ++s) {
        v2f bv = *(const v2f*)&ktc[nt * 16 + ml][4 * s + kg];
        acc = __builtin_amdgcn_wmma_f32_16x16x4_f32(
            /*neg_a=*/false, areg[s], /*neg_b=*/false, bv,
            /*c_mod=*/(short)0, acc, /*reuse_a=*/false, /*reuse_b=*/false);
      }
      const int   nl  = nt * 16 + ml;   // column within chunk
      const float mnv = mn_s[nl];
#pragma unroll
      for (int r = 0; r < 8; ++r) {
        const int mm = r + ((lane >> 4) << 3);  // C-layout: lanes 16-31 -> M=r+8
        d2buf[mm][nl] = hn_s[mm] + mnv - 2.0f * acc[r];
      }
    }
    __syncthreads();  // d2 tile complete; kt[buf] reads done (safe to refill)

    // streaming top-50 update (8 candidates per lane per chunk)
#pragma unroll
    for (int j = 0; j < NC / 8; ++j) {
      const int col = sslot + 8 * j;
      const float dv = d2buf[srow][col];
      if (dv < worst) {
        nd[worstpos] = dv;
        ni[worstpos] = c0 + col;
        worst = nd[0]; worstpos = 0;
        for (int i = 1; i < P_SZ; ++i)
          if (nd[i] > worst) { worst = nd[i]; worstpos = i; }
      }
    }
    buf ^= 1;
  }

  // ======================= merge + weighted reduce ========================
  // sort private list ascending
  for (int i = 1; i < P_SZ; ++i) {
    const float dv = nd[i]; const int iv = ni[i];
    int j = i - 1;
    while (j >= 0 && nd[j] > dv) { nd[j + 1] = nd[j]; ni[j + 1] = ni[j]; --j; }
    nd[j + 1] = dv; ni[j + 1] = iv;
  }

  // 8-way merge across the quarter-wave: exact global top-50 for this row
  const float* vals = mem_values + (size_t)a * C_SZ;
  const int l8 = lane & 7;
  int   hp   = 0;
  float wloc = 0.f, qloc = 0.f;
  for (int r = 0; r < P_SZ; ++r) {
    const float hd = (hp < P_SZ) ? nd[hp] : __builtin_inff();
    float mind = hd;
    mind = fminf(mind, __shfl_xor(mind, 4, 8));
    mind = fminf(mind, __shfl_xor(mind, 2, 8));
    mind = fminf(mind, __shfl_xor(mind, 1, 8));
    int key = (hd == mind) ? l8 : 8;
    key = imin(key, __shfl_xor(key, 4, 8));
    key = imin(key, __shfl_xor(key, 2, 8));
    key = imin(key, __shfl_xor(key, 1, 8));
    if (l8 == key) {
      const float wi = 1.0f / (hd + DELTA_K);
      wloc += wi;
      qloc = __builtin_fmaf(wi, vals[ni[hp]], qloc);
      ++hp;
    }
  }
  wloc += __shfl_xor(wloc, 4, 8);
  wloc += __shfl_xor(wloc, 2, 8);
  wloc += __shfl_xor(wloc, 1, 8);
  qloc += __shfl_xor(qloc, 4, 8);
  qloc += __shfl_xor(qloc, 2, 8);
  qloc += __shfl_xor(qloc, 1, 8);
  if (l8 == 0) out[(size_t)(brow0 + srow) * A_SZ + a] = qloc / wloc;
}

// ---------------------------------------------------------------------------
extern "C" void kernel_launch(void* const* d_in, const int* in_sizes, int n_in,
                              void* d_out, int out_size, void* d_ws, size_t ws_size,
                              hipStream_t stream) {
  (void)in_sizes; (void)n_in; (void)out_size; (void)ws_size;
  const float* obs  = (const float*)d_in[0];
  const float* W    = (const float*)d_in[1];
  const float* bias = (const float*)d_in[2];
  const float* keys = (const float*)d_in[3];
  const float* vals = (const float*)d_in[4];
  // d_in[5] is p (==50), baked in at compile time as P_SZ.
  float* out = (float*)d_out;

  float* h  = (float*)d_ws;                  // [B_SZ, D_SZ]
  float* hn = h + (size_t)B_SZ * D_SZ;       // [B_SZ]

  nec_embed<<<B_SZ, 128, 0, stream>>>(obs, W, bias, h, hn);

  dim3 grid(A_SZ, B_SZ / BT);
  nec_topk<<<grid, 128, 0, stream>>>(keys, vals, h, hn, out);
}